// FocalLoss_81999515615847
// MI455X (gfx1250) — compile-verified
//
#include <hip/hip_runtime.h>
#include <math.h>

// ---------------------------------------------------------------------------
// FocalLoss for rotated anchors, MI455X (gfx1250).
// Memory-bound: one streaming pass over 256MB of classifications (b128 loads),
// distance Gram-matrix via V_WMMA_F32_16X16X4_F32, fast native v_sqrt_f32,
// bit-packed argmin (v_min_u32), wave reductions, per-image float atomics.
// ---------------------------------------------------------------------------

typedef __attribute__((ext_vector_type(2))) float v2f;
typedef __attribute__((ext_vector_type(8))) float v8f;

#define NB 8
#define NA 100000
#define NC 80
#define NK 32
#define TILES_PER_IMG (NA / 16)                                   // 6250
#define WAVES_PER_BLK 8
#define BLKS_PER_IMG ((TILES_PER_IMG + WAVES_PER_BLK - 1) / WAVES_PER_BLK) // 782

__global__ void zero_ws_kernel(float* ws) {
    if (threadIdx.x < NB * 4) ws[threadIdx.x] = 0.0f;
}

__device__ __forceinline__ float fast_sqrtf(float x) {
    return __builtin_amdgcn_sqrtf(x);          // raw v_sqrt_f32, ~1 ulp
}
__device__ __forceinline__ float clampf(float x, float lo, float hi) {
    return __builtin_amdgcn_fmed3f(x, lo, hi); // single v_med3_f32
}

__global__ __launch_bounds__(256) void focal_main_kernel(
    const float* __restrict__ cls,   // [B,N,C]
    const float* __restrict__ reg,   // [B,N,3]
    const float* __restrict__ anc,   // [1,N,3]
    const float* __restrict__ ann,   // [B,K,4]
    float* __restrict__ ws)          // [B][cls_sum, xy_sum, ang_sum, n_pos]
{
    __shared__ float s_gx[NK], s_gy[NK], s_ga[NK], s_gsq[NK];
    __shared__ int   s_lab[NK];

    const int b   = blockIdx.x / BLKS_PER_IMG;
    const int tb  = blockIdx.x % BLKS_PER_IMG;
    const int tid = threadIdx.x;

    // Stage per-image annotations (32 x 4 floats) into LDS once per block.
    if (tid < NK) {
        const float* a = ann + ((size_t)b * NK + tid) * 4;
        float gx = a[0], gy = a[1], ga = a[2];
        s_gx[tid]  = gx;
        s_gy[tid]  = gy;
        s_ga[tid]  = ga;
        s_gsq[tid] = gx * gx + gy * gy;
        s_lab[tid] = (int)a[3];
    }
    __syncthreads();

    const int wave = tid >> 5;
    const int lane = tid & 31;
    const int tile = tb * WAVES_PER_BLK + wave;     // wave-uniform
    if (tile >= TILES_PER_IMG) return;              // uniform exit, EXEC stays full

    const int  n0  = tile * 16;
    const int  col = lane & 15;
    const bool hi  = lane >= 16;

    // Per-lane anchor (column of B-matrix). Halves duplicate -> L1 hits.
    const float* ap = anc + (size_t)(n0 + col) * 3;
    const float xa = ap[0], ya = ap[1], aa = ap[2];

    // Prefetch this tile's classification rows (16 anchors * 80 cls = 5120B).
    const float* crow = cls + ((size_t)b * NA + n0) * NC;
    __builtin_prefetch(crow + (size_t)lane * 40, 0, 0);

    // --- Phase 1: distance Gram matrix via WMMA f32 16x16x4 -----------------
    // D[m][n] = (xg_m^2+yg_m^2) + (xa_n^2+ya_n^2) - 2(xa_n*xg_m + ya_n*yg_m)
    //         = dx^2 + dy^2  for annotation m, anchor n.
    // A 16x4 layout: lanes0-15 {V0=K0,V1=K1}, lanes16-31 {V0=K2,V1=K3}.
    // B 4x16 layout: lanes0-15 rows {K0,K1}, lanes16-31 rows {K2,K3}.
    const float sqa = xa * xa + ya * ya;
    v2f bf;
    bf.x = hi ? 1.0f : -2.0f * xa;   // rows: [-2x, -2y, 1, x^2+y^2]
    bf.y = hi ? sqa  : -2.0f * ya;

    const int m0 = lane & 15;
    v2f a0, a1;                      // rows: [xg, yg, xg^2+yg^2, 1]
    a0.x = hi ? s_gsq[m0]      : s_gx[m0];
    a0.y = hi ? 1.0f           : s_gy[m0];
    a1.x = hi ? s_gsq[m0 + 16] : s_gx[m0 + 16];
    a1.y = hi ? 1.0f           : s_gy[m0 + 16];

    v8f zero8 = {0.f, 0.f, 0.f, 0.f, 0.f, 0.f, 0.f, 0.f};
    v8f d0 = __builtin_amdgcn_wmma_f32_16x16x4_f32(false, a0, false, bf,
                                                   (short)0, zero8, false, false);
    v8f d1 = __builtin_amdgcn_wmma_f32_16x16x4_f32(false, a1, false, bf,
                                                   (short)0, zero8, false, false);

    // Per-lane argmin over 16 annotations (this half) with a bit-packed key:
    // dist >= 0 so IEEE bits order monotonically; low 5 mantissa bits carry
    // the annotation index -> one v_min_u32 per candidate, ties -> lowest idx.
    unsigned key = 0xFFFFFFFFu;
    const int mbase = hi ? 8 : 0;
#pragma unroll
    for (int v = 0; v < 8; ++v) {
        const int m = mbase + v;
        const float dd = fmaxf(d0[v], 0.0f);
        const float dist = fmaf(10.0f, fast_sqrtf(dd),
                                __builtin_fabsf(aa - s_ga[m]));
        const unsigned k = (__float_as_uint(dist) & 0xFFFFFFE0u) | (unsigned)m;
        key = min(key, k);
    }
#pragma unroll
    for (int v = 0; v < 8; ++v) {
        const int m = mbase + v + 16;
        const float dd = fmaxf(d1[v], 0.0f);
        const float dist = fmaf(10.0f, fast_sqrtf(dd),
                                __builtin_fabsf(aa - s_ga[m]));
        const unsigned k = (__float_as_uint(dist) & 0xFFFFFFE0u) | (unsigned)m;
        key = min(key, k);
    }

    // Merge the two annotation halves (lane L <-> L^16 hold the same anchor).
    key = min(key, (unsigned)__shfl_xor((int)key, 16, 32));
    const float dmin = __uint_as_float(key & 0xFFFFFFE0u);
    const int   amin = (int)(key & 31u);

    const bool pos = dmin <= 11.0f * 50.0f;
    const bool neg = dmin >= 13.0f * 50.0f;
    const int  label = s_lab[amin];
    const unsigned pk = (pos ? 1u : 0u) | (neg ? 2u : 0u) | ((unsigned)label << 8);

    // --- Phase 2: focal classification loss, float4-vectorized --------------
    // 16 anchors * 80 classes = 1280 contiguous floats = 320 float4 chunks.
    // Each 16B chunk is 4-aligned and 80%4==0 -> never crosses an anchor row.
    const float4* crow4 = reinterpret_cast<const float4*>(crow);
    float clsSum = 0.0f;
#pragma unroll
    for (int it = 0; it < 10; ++it) {
        const int chunk = it * 32 + lane;
        const int base  = chunk * 4;
        const int j     = base / NC;           // anchor within tile
        const int c0    = base - j * NC;       // first class of chunk
        const unsigned pkj = __shfl(pk, j, 32);
        const float4 q = crow4[chunk];
        const float pv[4] = {q.x, q.y, q.z, q.w};
        const bool posj = (pkj & 1u) != 0u;
        const bool negj = (pkj & 2u) != 0u;
        const float keep = (posj || negj) ? 1.0f : 0.0f;   // 0 in ignore band
        const int lb = (int)(pkj >> 8);
#pragma unroll
        for (int e = 0; e < 4; ++e) {
            const float p = clampf(pv[e], 1.0e-4f, 1.0f - 1.0e-4f);
            const bool tpos = posj && ((c0 + e) == lb);
            const float pw  = tpos ? (1.0f - p) : p;
            const float af  = tpos ? 0.25f : 0.75f;
            const float bce = -__logf(tpos ? p : (1.0f - p));
            clsSum += keep * af * pw * pw * bce;
        }
    }

    // --- Phase 3: regression loss (each anchor counted once, lanes 0-15) ----
    const float* rp = reg + ((size_t)b * NA + (n0 + col)) * 3;
    const float rx = rp[0], ry = rp[1], ra = rp[2];
    const float tx = s_gx[amin] - xa;
    const float ty = s_gy[amin] - ya;
    const float ta = s_ga[amin] - aa;
    const float dxx = __builtin_fabsf(tx - rx);
    const float dyy = __builtin_fabsf(ty - ry);
    const float dang = 1.0f - __cosf(ta - ra);
    const float th  = 1.0f / 9.0f;
    const float lx = (dxx <= th) ? 4.5f * dxx * dxx : dxx - 0.5f / 9.0f;
    const float ly = (dyy <= th) ? 4.5f * dyy * dyy : dyy - 0.5f / 9.0f;
    const float act = (pos && !hi) ? 1.0f : 0.0f;
    float xySum  = act * (lx + ly);
    float angSum = act * dang;
    float posCnt = act;

    // --- Wave butterfly reduction, then 4 atomics per wave ------------------
#pragma unroll
    for (int s = 16; s > 0; s >>= 1) {
        clsSum += __shfl_xor(clsSum, s, 32);
        xySum  += __shfl_xor(xySum,  s, 32);
        angSum += __shfl_xor(angSum, s, 32);
        posCnt += __shfl_xor(posCnt, s, 32);
    }
    if (lane == 0) {
        atomicAdd(&ws[b * 4 + 0], clsSum);
        atomicAdd(&ws[b * 4 + 1], xySum);
        atomicAdd(&ws[b * 4 + 2], angSum);
        atomicAdd(&ws[b * 4 + 3], posCnt);
    }
}

__global__ void focal_final_kernel(const float* __restrict__ ws,
                                   float* __restrict__ out) {
    if (threadIdx.x == 0) {
        float cs = 0.0f, rs = 0.0f;
        for (int b = 0; b < NB; ++b) {
            const float np    = ws[b * 4 + 3];
            const float denom = fmaxf(np, 1.0f);
            cs += ws[b * 4 + 0] / denom;
            const float r = ws[b * 4 + 1] / (2.0f * denom) +
                            ws[b * 4 + 2] / denom;
            rs += (np > 0.0f) ? r : 0.0f;
        }
        out[0] = cs / (float)NB;
        out[1] = rs / (float)NB;
    }
}

extern "C" void kernel_launch(void* const* d_in, const int* in_sizes, int n_in,
                              void* d_out, int out_size, void* d_ws, size_t ws_size,
                              hipStream_t stream) {
    (void)in_sizes; (void)n_in; (void)out_size; (void)ws_size;
    const float* cls = (const float*)d_in[0];
    const float* reg = (const float*)d_in[1];
    const float* anc = (const float*)d_in[2];
    const float* ann = (const float*)d_in[3];
    float* ws  = (float*)d_ws;
    float* out = (float*)d_out;

    zero_ws_kernel<<<1, 64, 0, stream>>>(ws);
    focal_main_kernel<<<NB * BLKS_PER_IMG, 256, 0, stream>>>(cls, reg, anc, ann, ws);
    focal_final_kernel<<<1, 32, 0, stream>>>(ws, out);
}